// NCELoss_3882650436398
// MI455X (gfx1250) — compile-verified
//
#include <hip/hip_runtime.h>
#include <hip/hip_bf16.h>
#include <math.h>

// ---------------------------------------------------------------------------
// NCE loss for MI455X (gfx1250, wave32).
//   logits[n][k] = dot(input[n,:], weight[idx[n][k],:]) + bias[idx[n][k]]
// computed with fp32 WMMA (V_WMMA_F32_16X16X4_F32), 16 samples per wave,
// diagonal extraction of the 16x16 accumulator.
// ---------------------------------------------------------------------------

typedef __attribute__((ext_vector_type(2))) float v2f;
typedef __attribute__((ext_vector_type(8))) float v8f;

#define N_SAMPLES 32768
#define EDIM      256
#define KNOISE    10
#define KTOT      11          // 1 target + 10 noise
#define NORM_TERM_F 9.0f
#define KF        10.0f       // float(NOISE_RATIO)

#define WAVES_PER_BLOCK   8
#define SAMPLES_PER_WAVE  16
#define SAMPLES_PER_BLOCK (WAVES_PER_BLOCK * SAMPLES_PER_WAVE)   // 128
#define NBLOCKS           (N_SAMPLES / SAMPLES_PER_BLOCK)        // 256

__global__ __launch_bounds__(256) void nce_partial_kernel(
    const float* __restrict__ input,          // [N, E]
    const int*   __restrict__ target,         // [N]
    const int*   __restrict__ noise_samples,  // [N, KNOISE]
    const float* __restrict__ noise,          // [V]
    const float* __restrict__ weight,         // [V, E]
    const float* __restrict__ bias,           // [V]
    float*       __restrict__ partial)        // [NBLOCKS]
{
    __shared__ float lds_logits[WAVES_PER_BLOCK][SAMPLES_PER_WAVE][KTOT];
    __shared__ float lds_wsum[WAVES_PER_BLOCK];

    const int tid  = threadIdx.x;
    const int wave = tid >> 5;
    const int lane = tid & 31;
    const int m    = lane & 15;   // sample-in-tile == B column for this lane
    const int hh   = lane >> 4;   // which K-half of the 16x4 / 4x16 operand
    const int base = (blockIdx.x * WAVES_PER_BLOCK + wave) * SAMPLES_PER_WAVE;
    const int s    = base + m;    // this lane's sample

    // A operand: lane holds input[s][kk + 2*hh .. kk + 2*hh + 1] per K-chunk.
    const float* arow = input + s * EDIM + 2 * hh;

    for (int k = 0; k < KTOT; ++k) {
        const int row = (k == 0) ? target[s]
                                 : noise_samples[s * KNOISE + (k - 1)];
        const float* brow = weight + row * EDIM + 2 * hh;

        // Prefetch next k-slot's gathered weight row into cache.
        if (k + 1 < KTOT) {
            const int nrow = noise_samples[s * KNOISE + k];
            __builtin_prefetch(weight + nrow * EDIM, 0, 3);
        }

        v8f acc = {0.f, 0.f, 0.f, 0.f, 0.f, 0.f, 0.f, 0.f};
        #pragma unroll 8
        for (int kk = 0; kk < EDIM; kk += 4) {
            v2f a = *(const v2f*)(arow + kk);   // A 16x4 fp32 slice
            v2f b = *(const v2f*)(brow + kk);   // B 4x16 fp32 slice (gathered)
            // D = A*B + C, fp32 exact:  v_wmma_f32_16x16x4_f32
            acc = __builtin_amdgcn_wmma_f32_16x16x4_f32(
                false, a, false, b, (short)0, acc, false, false);
        }

        // Diagonal of C is logits for the 16 samples of this tile:
        //   VGPR r, lane r      -> M = r      (r = 0..7)
        //   VGPR r, lane r + 24 -> M = r + 8
        if (lane < 8) {
            lds_logits[wave][lane][k] = acc[lane] + bias[row];
        } else if (lane >= 24) {
            lds_logits[wave][lane - 16][k] = acc[lane - 24] + bias[row];
        }
    }
    __syncthreads();

    // ---- epilogue: lanes 0..15 each own one sample --------------------------
    float contrib = 0.0f;
    if (lane < 16) {
        const float dlogit = lds_logits[wave][m][0];
        const float dprob  = expf(dlogit - NORM_TERM_F);
        const float nt     = noise[target[s]];
        const float rnn    = logf(dprob / (dprob + KF * nt));

        float nloss = 0.0f;
        #pragma unroll
        for (int k = 1; k < KTOT; ++k) {
            const float p  = expf(lds_logits[wave][m][k] - NORM_TERM_F);
            const float np = noise[noise_samples[s * KNOISE + (k - 1)]];
            nloss += logf((KF * np) / (p + KF * np));
        }
        contrib = -(rnn + nloss);
    }

    // deterministic wave reduction (lanes 16..31 contribute 0)
    #pragma unroll
    for (int off = 16; off >= 1; off >>= 1)
        contrib += __shfl_down(contrib, off, 32);
    if (lane == 0) lds_wsum[wave] = contrib;
    __syncthreads();

    if (tid == 0) {
        float bsum = 0.0f;
        #pragma unroll
        for (int w = 0; w < WAVES_PER_BLOCK; ++w) bsum += lds_wsum[w];
        partial[blockIdx.x] = bsum;
    }
}

__global__ __launch_bounds__(256) void nce_reduce_kernel(
    const float* __restrict__ partial, float* __restrict__ out)
{
    __shared__ float red[NBLOCKS];
    const int t = threadIdx.x;
    red[t] = partial[t];
    __syncthreads();
    #pragma unroll
    for (int off = NBLOCKS / 2; off > 0; off >>= 1) {
        if (t < off) red[t] += red[t + off];
        __syncthreads();
    }
    if (t == 0) out[0] = red[0] / (float)N_SAMPLES;
}

extern "C" void kernel_launch(void* const* d_in, const int* in_sizes, int n_in,
                              void* d_out, int out_size, void* d_ws, size_t ws_size,
                              hipStream_t stream)
{
    const float* input         = (const float*)d_in[0];
    const int*   target        = (const int*)  d_in[1];
    const int*   noise_samples = (const int*)  d_in[2];
    const float* noise         = (const float*)d_in[3];
    const float* weight        = (const float*)d_in[4];
    const float* bias          = (const float*)d_in[5];
    float*       out           = (float*)d_out;
    float*       partial       = (float*)d_ws;   // NBLOCKS floats of scratch

    nce_partial_kernel<<<NBLOCKS, 256, 0, stream>>>(
        input, target, noise_samples, noise, weight, bias, partial);
    nce_reduce_kernel<<<1, NBLOCKS, 0, stream>>>(partial, out);
}